// ResidualGNNsWithInputAttention_66975720013951
// MI455X (gfx1250) — compile-verified
//
#include <hip/hip_runtime.h>
#include <math.h>

typedef __attribute__((ext_vector_type(2))) float v2f;
typedef __attribute__((ext_vector_type(8))) float v8f;

#define NG_     256
#define NPG_    64
#define NF_     64
#define NODES_  (NG_*NPG_)      // 16384
#define EPER_   1024
#define NEDGE_  (NG_*EPER_)     // 262144
#define HCP_    512             // padded hidden (504 -> 512)
#define TRIU_   2016
#define D1_     4032
#define HID_    1024

// ---------------------------------------------------------------------------
// fp32 WMMA GEMM:  C[M,N] = A[M,K] @ B[K,N] (+ bias[N]), row-major.
// M,N multiples of 64; K multiple of 4. Batched via grid.y + strides.
// Each wave computes a 64x64 macro-tile as 4x4 WMMA 16x16 tiles
// (V_WMMA_F32_16X16X4_F32, full fp32). Per K-step: 4 A-frags + 4 B-frags
// (2KB/wave) feed 16 WMMAs (32768 FLOP) -> ~16 FLOP/byte from L2.
// Block = 128 threads = 4 waves = 1 wave/SIMD; launch_bounds(128,1) gives
// each wave the full VGPR budget so the 16 v8f accumulators never spill.
// ---------------------------------------------------------------------------
__global__ void __launch_bounds__(128, 1)
wmma_gemm_f32(const float* __restrict__ A, int lda, long sA,
              const float* __restrict__ B, int ldb, long sB,
              float* __restrict__ C, int ldc, long sC,
              const float* __restrict__ bias,
              int M, int N, int K)
{
    const int b = blockIdx.y;
    A += sA * (long)b; B += sB * (long)b; C += sC * (long)b;

    const int lane   = threadIdx.x & 31;
    const int wave   = threadIdx.x >> 5;
    const int tilesN = N >> 6;                         // 64-wide macro tiles
    const int totT   = (M >> 6) * tilesN;
    const int tile   = blockIdx.x * (blockDim.x >> 5) + wave;
    if (tile >= totT) return;                          // wave-uniform: EXEC all-1s

    const int m0   = (tile / tilesN) << 6;
    const int n0   = (tile % tilesN) << 6;
    const int half = lane >> 4;                        // 0: K pair {0,1}, 1: {2,3}
    const int l    = lane & 15;

    v8f acc[4][4] = {};
    // A-frag base: lane(l,half) -> A[m0+mi*16+l][k+2*half .. +1]
    const float* Ab = A + (long)(m0 + l) * lda + (half << 1);
    // B-frag base: lane(l,half) -> B[k+2*half][n0+ni*16+l] (and +ldb)
    const float* Bb = B + (long)(half << 1) * ldb + (n0 + l);

    for (int k = 0; k < K; k += 4) {
        if ((k & 63) == 0 && k + 64 < K) {             // uniform-branch prefetch
            __builtin_prefetch(Ab + (k + 64), 0, 3);   // -> global_prefetch_b8
            __builtin_prefetch(Bb + (long)(k + 64) * ldb, 0, 3);
        }
        v2f a[4], bb[4];
#pragma unroll
        for (int mi = 0; mi < 4; ++mi)
            a[mi] = *(const v2f*)(Ab + (long)(mi << 4) * lda + k);
#pragma unroll
        for (int ni = 0; ni < 4; ++ni) {
            bb[ni].x = Bb[(long)k * ldb + (ni << 4)];
            bb[ni].y = Bb[(long)k * ldb + ldb + (ni << 4)];
        }
#pragma unroll
        for (int mi = 0; mi < 4; ++mi)
#pragma unroll
            for (int ni = 0; ni < 4; ++ni)
                acc[mi][ni] = __builtin_amdgcn_wmma_f32_16x16x4_f32(
                    false, a[mi], false, bb[ni], (short)0, acc[mi][ni], false, false);
    }

    float bv[4];
#pragma unroll
    for (int ni = 0; ni < 4; ++ni)
        bv[ni] = bias ? bias[n0 + (ni << 4) + l] : 0.0f;

#pragma unroll
    for (int mi = 0; mi < 4; ++mi) {
        // D-frag: VGPR r -> row m0+mi*16 + r + 8*half, col n0+ni*16 + l
        float* Cp = C + (long)(m0 + (mi << 4) + (half << 3)) * ldc + (n0 + l);
#pragma unroll
        for (int ni = 0; ni < 4; ++ni)
#pragma unroll
            for (int r = 0; r < 8; ++r)
                Cp[(long)r * ldc + (ni << 4)] = acc[mi][ni][r] + bv[ni];
    }
}

// ---------------------------------------------------------------------------
// Degree / dinv
// ---------------------------------------------------------------------------
__global__ void deg_init_k(float* deg) {
    int n = blockIdx.x * blockDim.x + threadIdx.x;
    if (n < NODES_) deg[n] = 1.0f;            // +1 self loop
}
__global__ void deg_acc_k(const int* __restrict__ dst, float* deg) {
    int e = blockIdx.x * blockDim.x + threadIdx.x;
    if (e < NEDGE_) atomicAdd(&deg[dst[e]], 1.0f);
}
__global__ void dinv_k(const float* __restrict__ deg, float* dinv) {
    int n = blockIdx.x * blockDim.x + threadIdx.x;
    if (n < NODES_) dinv[n] = rsqrtf(deg[n]);
}

// ---------------------------------------------------------------------------
// Per-graph dense normalized adjacency: Adj[g][d][s] = dinv_d*dinv_s*mult(s->d)
//                                       + (d==s ? dinv_d^2 : 0)
// ---------------------------------------------------------------------------
__global__ void adj_build_k(const int* __restrict__ src, const int* __restrict__ dst,
                            const float* __restrict__ dinv, float* __restrict__ adj)
{
    const int g = blockIdx.x, t = threadIdx.x;
    __shared__ float As[NPG_ * NPG_];
    __shared__ float ds[NPG_];
    for (int i = t; i < NPG_ * NPG_; i += blockDim.x) As[i] = 0.0f;
    if (t < NPG_) ds[t] = dinv[g * NPG_ + t];
    __syncthreads();
    if (t < NPG_) As[t * NPG_ + t] = ds[t] * ds[t];   // self-loop term
    __syncthreads();
    for (int e = t; e < EPER_; e += blockDim.x) {
        int s = src[g * EPER_ + e] & (NPG_ - 1);
        int d = dst[g * EPER_ + e] & (NPG_ - 1);
        atomicAdd(&As[d * NPG_ + s], ds[s] * ds[d]);
    }
    __syncthreads();
    for (int i = t; i < NPG_ * NPG_; i += blockDim.x) adj[(long)g * NPG_ * NPG_ + i] = As[i];
}

// ---------------------------------------------------------------------------
// Zero-pad a [r,c] matrix into [R,C]
// ---------------------------------------------------------------------------
__global__ void pad_mat_k(const float* __restrict__ src, int r, int c,
                          float* __restrict__ dst, int R, int C)
{
    long i = (long)blockIdx.x * blockDim.x + threadIdx.x;
    if (i >= (long)R * C) return;
    int row = (int)(i / C), col = (int)(i % C);
    dst[i] = (row < r && col < c) ? src[(long)row * c + col] : 0.0f;
}

// ---------------------------------------------------------------------------
// global_mean_pool -> hcat[g][layer*504 + c]
// ---------------------------------------------------------------------------
__global__ void pool_k(const float* __restrict__ H, float* __restrict__ hcat, int layer)
{
    int g = blockIdx.x, c = threadIdx.x;     // block = 512
    if (c >= 504) return;
    float s = 0.0f;
    const float* base = H + (long)g * NPG_ * HCP_ + c;
    for (int r = 0; r < NPG_; ++r) s += base[r * HCP_];
    hcat[(long)g * TRIU_ + layer * 504 + c] = s * (1.0f / NPG_);
}

// ---------------------------------------------------------------------------
// BatchNorm1d (training mode, biased var) over rows=256; one block per column.
// ---------------------------------------------------------------------------
__global__ void bn_k(const float* __restrict__ X, float* __restrict__ Y, int ld,
                     const float* __restrict__ gamma, const float* __restrict__ beta,
                     int relu)
{
    const int col = blockIdx.x, t = threadIdx.x;   // blockDim == rows == 256
    float v = X[(long)t * ld + col];
    __shared__ float ss[256], sq[256];
    ss[t] = v; sq[t] = v * v;
    __syncthreads();
    for (int s = 128; s > 0; s >>= 1) {
        if (t < s) { ss[t] += ss[t + s]; sq[t] += sq[t + s]; }
        __syncthreads();
    }
    __shared__ float mv[2];
    if (t == 0) {
        float m = ss[0] * (1.0f / 256.0f);
        mv[0] = m; mv[1] = sq[0] * (1.0f / 256.0f) - m * m;
    }
    __syncthreads();
    float y = (v - mv[0]) * rsqrtf(mv[1] + 1e-5f) * gamma[col] + beta[col];
    if (relu) y = fmaxf(y, 0.0f);
    Y[(long)t * ld + col] = y;
}

// ---------------------------------------------------------------------------
// Strict-upper-triangle index table + extraction
// ---------------------------------------------------------------------------
__global__ void triu_idx_k(int2* ij)
{
    int t = blockIdx.x * blockDim.x + threadIdx.x;
    if (t >= TRIU_) return;
    int rem = t, i = 0;
    while (i < NF_ - 1) { int cnt = NF_ - 1 - i; if (rem < cnt) break; rem -= cnt; ++i; }
    ij[t] = make_int2(i, i + 1 + rem);
}
__global__ void xv_k(const float* __restrict__ x, const int2* __restrict__ ij,
                     float* __restrict__ xv)
{
    int g = blockIdx.x;
    for (int t = threadIdx.x; t < TRIU_; t += blockDim.x) {
        int2 p = ij[t];
        xv[(long)g * TRIU_ + t] = x[((long)g * NPG_ + p.x) * NF_ + p.y];
    }
}

// ---------------------------------------------------------------------------
// att[g] = sigmoid(dot(xv_bn[g], Watt) + batt)
// ---------------------------------------------------------------------------
__global__ void att_k(const float* __restrict__ xv, const float* __restrict__ Watt,
                      const float* __restrict__ batt, float* __restrict__ att)
{
    int g = blockIdx.x, t = threadIdx.x;
    float a = 0.0f;
    for (int c = t; c < TRIU_; c += 256) a += xv[(long)g * TRIU_ + c] * Watt[c];
    __shared__ float ss[256];
    ss[t] = a; __syncthreads();
    for (int s = 128; s > 0; s >>= 1) { if (t < s) ss[t] += ss[t + s]; __syncthreads(); }
    if (t == 0) att[g] = 1.0f / (1.0f + expf(-(ss[0] + batt[0])));
}

// z[g] = [ att*xv + (1-att)*h*0.5 , h ]
__global__ void fuse_k(const float* __restrict__ xv, const float* __restrict__ h,
                       const float* __restrict__ att, float* __restrict__ z)
{
    int g = blockIdx.x;
    float a = att[g];
    for (int c = threadIdx.x; c < TRIU_; c += blockDim.x) {
        float xvv = xv[(long)g * TRIU_ + c];
        float hv  = h[(long)g * TRIU_ + c];
        z[(long)g * D1_ + c]          = a * xvv + (1.0f - a) * hv * 0.5f;
        z[(long)g * D1_ + TRIU_ + c]  = hv;
    }
}

// final 512 -> 2 head
__global__ void head_k(const float* __restrict__ z3, const float* __restrict__ W,
                       const float* __restrict__ b, float* __restrict__ out)
{
    int g = blockIdx.x, t = threadIdx.x;
    float a0 = 0.0f, a1 = 0.0f;
    for (int k = t; k < 512; k += 256) {
        float v = z3[(long)g * 512 + k];
        a0 += v * W[k * 2 + 0];
        a1 += v * W[k * 2 + 1];
    }
    __shared__ float s0[256], s1[256];
    s0[t] = a0; s1[t] = a1; __syncthreads();
    for (int s = 128; s > 0; s >>= 1) {
        if (t < s) { s0[t] += s0[t + s]; s1[t] += s1[t + s]; }
        __syncthreads();
    }
    if (t == 0) { out[g * 2 + 0] = s0[0] + b[0]; out[g * 2 + 1] = s1[0] + b[1]; }
}

// ---------------------------------------------------------------------------
extern "C" void kernel_launch(void* const* d_in, const int* in_sizes, int n_in,
                              void* d_out, int out_size, void* d_ws, size_t ws_size,
                              hipStream_t stream)
{
    const float* x       = (const float*)d_in[0];
    const float* Wg0     = (const float*)d_in[1];
    const float* bg0     = (const float*)d_in[2];
    const float* Wg_rest = (const float*)d_in[3];
    const float* bg_rest = (const float*)d_in[4];
    const float* gx      = (const float*)d_in[5];
    const float* bx      = (const float*)d_in[6];
    const float* gh      = (const float*)d_in[7];
    const float* bh      = (const float*)d_in[8];
    const float* Watt    = (const float*)d_in[9];
    const float* batt    = (const float*)d_in[10];
    const float* Wm1     = (const float*)d_in[11];
    const float* bm1     = (const float*)d_in[12];
    const float* gm1     = (const float*)d_in[13];
    const float* bem1    = (const float*)d_in[14];
    const float* Wm2     = (const float*)d_in[15];
    const float* bm2     = (const float*)d_in[16];
    const float* gm2     = (const float*)d_in[17];
    const float* bem2    = (const float*)d_in[18];
    const float* Wm3     = (const float*)d_in[19];
    const float* bm3     = (const float*)d_in[20];
    const float* gm3     = (const float*)d_in[21];
    const float* bem3    = (const float*)d_in[22];
    const float* Wm4     = (const float*)d_in[23];
    const float* bm4     = (const float*)d_in[24];
    const int*   esrc    = (const int*)d_in[25];
    const int*   edst    = (const int*)d_in[26];
    float* out = (float*)d_out;

    // ---- carve workspace ----
    size_t off = 0;
    auto alloc = [&](size_t nfloats) -> float* {
        float* p = (float*)((char*)d_ws + off);
        off += ((nfloats * 4 + 255) / 256) * 256;
        return p;
    };
    float* deg  = alloc(NODES_);
    float* dinv = alloc(NODES_);
    float* adj  = alloc((size_t)NG_ * NPG_ * NPG_);
    float* Wp0  = alloc((size_t)NF_ * HCP_);
    float* Wp   = alloc((size_t)3 * HCP_ * HCP_);
    float* bp   = alloc((size_t)4 * HCP_);
    float* Htmp = alloc((size_t)NODES_ * HCP_);
    float* Hcur = alloc((size_t)NODES_ * HCP_);
    float* hcat = alloc((size_t)NG_ * TRIU_);
    float* xv   = alloc((size_t)NG_ * TRIU_);
    float* att  = alloc(NG_);
    float* z    = alloc((size_t)NG_ * D1_);
    float* z1   = alloc((size_t)NG_ * HID_);
    float* z2   = alloc((size_t)NG_ * 512);
    float* z3   = alloc((size_t)NG_ * 512);
    int2*  ij   = (int2*)alloc(TRIU_ * 2);
    (void)ws_size; (void)in_sizes; (void)n_in; (void)out_size;

    auto gemm = [&](const float* A, int lda, long sA, const float* B, int ldb, long sB,
                    float* C, int ldc, long sC, const float* bias,
                    int M, int N, int K, int batch) {
        int tiles = (M / 64) * (N / 64);               // 64x64 macro-tiles per wave
        dim3 grid((tiles + 3) / 4, batch);             // 4 waves (128 thr) per block
        hipLaunchKernelGGL(wmma_gemm_f32, grid, dim3(128), 0, stream,
                           A, lda, sA, B, ldb, sB, C, ldc, sC, bias, M, N, K);
    };

    // ---- graph structure: degrees, dinv, dense normalized adjacency ----
    hipLaunchKernelGGL(deg_init_k, dim3(NODES_ / 256), dim3(256), 0, stream, deg);
    hipLaunchKernelGGL(deg_acc_k, dim3(NEDGE_ / 256), dim3(256), 0, stream, edst, deg);
    hipLaunchKernelGGL(dinv_k, dim3(NODES_ / 256), dim3(256), 0, stream, deg, dinv);
    hipLaunchKernelGGL(adj_build_k, dim3(NG_), dim3(256), 0, stream, esrc, edst, dinv, adj);

    // ---- pad weights/biases 504 -> 512 ----
    hipLaunchKernelGGL(pad_mat_k, dim3((NF_ * HCP_ + 255) / 256), dim3(256), 0, stream,
                       Wg0, NF_, 504, Wp0, NF_, HCP_);
    for (int i = 0; i < 3; ++i)
        hipLaunchKernelGGL(pad_mat_k, dim3((HCP_ * HCP_ + 255) / 256), dim3(256), 0, stream,
                           Wg_rest + (size_t)i * 504 * 504, 504, 504,
                           Wp + (size_t)i * HCP_ * HCP_, HCP_, HCP_);
    hipLaunchKernelGGL(pad_mat_k, dim3(2), dim3(256), 0, stream, bg0, 1, 504, bp, 1, HCP_);
    for (int i = 0; i < 3; ++i)
        hipLaunchKernelGGL(pad_mat_k, dim3(2), dim3(256), 0, stream,
                           bg_rest + (size_t)i * 504, 1, 504, bp + (size_t)(i + 1) * HCP_, 1, HCP_);

    hipLaunchKernelGGL(triu_idx_k, dim3(8), dim3(256), 0, stream, ij);

    // ---- GCN layer 0: Htmp = x @ Wp0 ; Hcur = Adj_g @ Htmp + b0 ; pool ----
    gemm(x, NF_, 0, Wp0, HCP_, 0, Htmp, HCP_, 0, nullptr, NODES_, HCP_, NF_, 1);
    gemm(adj, NPG_, (long)NPG_ * NPG_, Htmp, HCP_, (long)NPG_ * HCP_,
         Hcur, HCP_, (long)NPG_ * HCP_, bp, NPG_, HCP_, NPG_, NG_);
    hipLaunchKernelGGL(pool_k, dim3(NG_), dim3(512), 0, stream, Hcur, hcat, 0);

    // ---- GCN layers 1..3 ----
    for (int i = 0; i < 3; ++i) {
        gemm(Hcur, HCP_, 0, Wp + (size_t)i * HCP_ * HCP_, HCP_, 0,
             Htmp, HCP_, 0, nullptr, NODES_, HCP_, HCP_, 1);
        gemm(adj, NPG_, (long)NPG_ * NPG_, Htmp, HCP_, (long)NPG_ * HCP_,
             Hcur, HCP_, (long)NPG_ * HCP_, bp + (size_t)(i + 1) * HCP_,
             NPG_, HCP_, NPG_, NG_);
        hipLaunchKernelGGL(pool_k, dim3(NG_), dim3(512), 0, stream, Hcur, hcat, i + 1);
    }

    // ---- BN(h), triu extract + BN(xv), attention, fusion ----
    hipLaunchKernelGGL(bn_k, dim3(TRIU_), dim3(256), 0, stream, hcat, hcat, TRIU_, gh, bh, 0);
    hipLaunchKernelGGL(xv_k, dim3(NG_), dim3(256), 0, stream, x, ij, xv);
    hipLaunchKernelGGL(bn_k, dim3(TRIU_), dim3(256), 0, stream, xv, xv, TRIU_, gx, bx, 0);
    hipLaunchKernelGGL(att_k, dim3(NG_), dim3(256), 0, stream, xv, Watt, batt, att);
    hipLaunchKernelGGL(fuse_k, dim3(NG_), dim3(256), 0, stream, xv, hcat, att, z);

    // ---- MLP head ----
    gemm(z, D1_, 0, Wm1, HID_, 0, z1, HID_, 0, bm1, NG_, HID_, D1_, 1);
    hipLaunchKernelGGL(bn_k, dim3(HID_), dim3(256), 0, stream, z1, z1, HID_, gm1, bem1, 1);
    gemm(z1, HID_, 0, Wm2, 512, 0, z2, 512, 0, bm2, NG_, 512, HID_, 1);
    hipLaunchKernelGGL(bn_k, dim3(512), dim3(256), 0, stream, z2, z2, 512, gm2, bem2, 1);
    gemm(z2, 512, 0, Wm3, 512, 0, z3, 512, 0, bm3, NG_, 512, 512, 1);
    hipLaunchKernelGGL(bn_k, dim3(512), dim3(256), 0, stream, z3, z3, 512, gm3, bem3, 1);
    hipLaunchKernelGGL(head_k, dim3(NG_), dim3(256), 0, stream, z3, Wm4, bm4, out);
}